// DynGraphSAGE_87763361727277
// MI455X (gfx1250) — compile-verified
//
#include <hip/hip_runtime.h>
#include <limits.h>

#define N_NODES  100000
#define FDIM     128
#define LOOKBACK 3
#define N_EDGES  640000

typedef float v2f __attribute__((ext_vector_type(2)));
typedef float v8f __attribute__((ext_vector_type(8)));

// ---------------------------------------------------------------------------
// fill int buffer (reset segment-max accumulator to INT_MIN sentinel)
// ---------------------------------------------------------------------------
__global__ void dgs_fill_i32(int* __restrict__ p, long long n, int val) {
    long long i = (long long)blockIdx.x * blockDim.x + threadIdx.x;
    long long stride = (long long)gridDim.x * blockDim.x;
    for (; i < n; i += stride) p[i] = val;
}

// ---------------------------------------------------------------------------
// segment_max scatter: m is relu output (all >= 0), so IEEE bits are
// monotonic under signed-int compare. One wave per edge, int4 per lane.
// ---------------------------------------------------------------------------
__global__ void dgs_scatter_max(const float* __restrict__ m,
                                const int* __restrict__ src,
                                const int* __restrict__ dst,
                                int* __restrict__ agg) {
    int t = blockIdx.x * blockDim.x + threadIdx.x;
    int e = t >> 5;
    int lane = t & 31;
    if (e >= N_EDGES) return;
    int s = src[e];
    int d = dst[e];
    const int4* vp = (const int4*)(m + (long long)s * FDIM) + lane;
    int4 v = *vp;
    int* base = agg + (long long)d * FDIM + lane * 4;
    atomicMax(base + 0, v.x);
    atomicMax(base + 1, v.y);
    atomicMax(base + 2, v.z);
    atomicMax(base + 3, v.w);
}

// ---------------------------------------------------------------------------
// WMMA GEMM: Y[nrows,128] = X0 @ W0^T (+ X1 @ W1^T) + bias, optional relu.
// X1 phase decodes segment-max int bits: <0 (sentinel) => 0.0f.
// Block = 256 threads = 8 waves; wave w owns rows [blk*128 + w*16, +16).
// W staged in LDS pre-swizzled into V_WMMA_F32_16X16X4_F32 B-fragment order:
//   slot(nb,ks,lane) = {W[nb*16+nl][ks*4+2*hi], W[..][ks*4+2*hi+1]}
// A rows are clamped (not predicated): out-of-range rows only affect their
// own D rows, which the guarded store discards.
// ---------------------------------------------------------------------------
__device__ __forceinline__ void dgs_stage_w(const float* __restrict__ W, float* lws) {
    for (int idx = threadIdx.x; idx < FDIM * FDIM; idx += blockDim.x) {
        int n  = idx >> 7;
        int k  = idx & 127;
        int nb = n >> 4, nl = n & 15;
        int ks = k >> 2, r  = k & 3;
        int hi = r >> 1, j  = r & 1;
        int lane = (hi << 4) | nl;
        lws[((((nb << 5) + ks) << 5 | lane) << 1) | j] = W[idx];
    }
}

template <bool DECODE>
__device__ __forceinline__ v2f dgs_frag_a(const float* __restrict__ rowp, int ks) {
    v2f a;
    if (DECODE) {
        int2 iv = *(const int2*)((const int*)rowp + ks * 4);
        a.x = __int_as_float(iv.x < 0 ? 0 : iv.x);
        a.y = __int_as_float(iv.y < 0 ? 0 : iv.y);
    } else {
        a = *(const v2f*)(rowp + ks * 4);
    }
    return a;
}

template <bool DECODE>
__device__ __forceinline__ void dgs_gemm_phase(const float* __restrict__ X,
                                               const float* lws, int mAc,
                                               int hi, int lane, v8f acc[8]) {
    // per-lane base: row mAc, column offset 2*hi; fragment ks reads +ks*4 floats
    const float* rowp = X + (long long)mAc * FDIM + 2 * hi;
    v2f a = dgs_frag_a<DECODE>(rowp, 0);
#pragma unroll
    for (int ks = 0; ks < 32; ++ks) {
        // prefetch next A fragment so its wait hides behind this step's WMMAs
        v2f an = dgs_frag_a<DECODE>(rowp, (ks < 31) ? (ks + 1) : 31);
#pragma unroll
        for (int nb = 0; nb < 8; ++nb) {
            v2f b = *(const v2f*)(lws + ((((nb << 5) + ks) << 5 | lane) << 1));
            acc[nb] = __builtin_amdgcn_wmma_f32_16x16x4_f32(
                false, a, false, b, (short)0, acc[nb], false, false);
        }
        a = an;
    }
}

__global__ void __launch_bounds__(256)
dgs_gemm128(const float* __restrict__ X0, const float* __restrict__ W0,
            const float* __restrict__ X1, const float* __restrict__ W1,
            const float* __restrict__ bias, int do_relu,
            float* __restrict__ Y, int nrows) {
    __shared__ float lws[FDIM * FDIM];   // 64 KB

    int w    = threadIdx.x >> 5;
    int lane = threadIdx.x & 31;
    int nl   = lane & 15;
    int hi   = lane >> 4;
    int m0   = blockIdx.x * 128 + w * 16;
    int mA   = m0 + nl;
    int mAc  = min(mA, nrows - 1);       // clamped, unconditional loads

    v8f acc[8];
#pragma unroll
    for (int nb = 0; nb < 8; ++nb) acc[nb] = (v8f){0.f,0.f,0.f,0.f,0.f,0.f,0.f,0.f};

    dgs_stage_w(W0, lws);
    __syncthreads();
    dgs_gemm_phase<false>(X0, lws, mAc, hi, lane, acc);

    if (X1 != nullptr) {                 // uniform branch
        __syncthreads();
        dgs_stage_w(W1, lws);
        __syncthreads();
        dgs_gemm_phase<true>(X1, lws, mAc, hi, lane, acc);
    }

    // D layout: VGPR r -> row m0 + r + 8*hi, col nb*16 + nl
#pragma unroll
    for (int nb = 0; nb < 8; ++nb) {
        int col = nb * 16 + nl;
        float bv = bias ? bias[col] : 0.f;
#pragma unroll
        for (int r = 0; r < 8; ++r) {
            int row = m0 + r + 8 * hi;
            if (row < nrows) {
                float v = acc[nb][r] + bv;
                if (do_relu) v = fmaxf(v, 0.f);
                Y[(long long)row * FDIM + col] = v;
            }
        }
    }
}

// ---------------------------------------------------------------------------
// BatchNorm (training-mode batch stats) over axis 0
// ---------------------------------------------------------------------------
#define BN_ROWS 256
__global__ void dgs_bn_partial(const float* __restrict__ h, float* __restrict__ part,
                               int nrows) {
    int f  = threadIdx.x;                 // 128 threads
    int r0 = blockIdx.x * BN_ROWS;
    int re = min(r0 + BN_ROWS, nrows);
    float s = 0.f, q = 0.f;
    for (int r = r0; r < re; ++r) {
        float v = h[(long long)r * FDIM + f];
        s += v; q += v * v;
    }
    part[blockIdx.x * 256 + f]       = s;
    part[blockIdx.x * 256 + 128 + f] = q;
}

__global__ void dgs_bn_finalize(const float* __restrict__ part, int nblocks,
                                const float* __restrict__ gamma,
                                const float* __restrict__ beta,
                                float* __restrict__ sc) {
    int f = threadIdx.x;                  // 128 threads
    float s = 0.f, q = 0.f;
    for (int b = 0; b < nblocks; ++b) {
        s += part[b * 256 + f];
        q += part[b * 256 + 128 + f];
    }
    float inv_n = 1.0f / (float)N_NODES;
    float mu  = s * inv_n;
    float var = q * inv_n - mu * mu;
    float scale = gamma[f] * rsqrtf(var + 1e-5f);
    sc[f]       = scale;
    sc[128 + f] = beta[f] - mu * scale;
}

__global__ void dgs_bn_apply_relu(float* __restrict__ h, const float* __restrict__ sc,
                                  long long n) {
    long long i = (long long)blockIdx.x * blockDim.x + threadIdx.x;
    long long stride = (long long)gridDim.x * blockDim.x;
    for (; i < n; i += stride) {
        int f = (int)(i & 127);
        h[i] = fmaxf(h[i] * sc[f] + sc[128 + f], 0.f);
    }
}

// ---------------------------------------------------------------------------
extern "C" void kernel_launch(void* const* d_in, const int* in_sizes, int n_in,
                              void* d_out, int out_size, void* d_ws, size_t ws_size,
                              hipStream_t stream) {
    (void)in_sizes; (void)n_in; (void)out_size; (void)ws_size;

    const float* feat  = (const float*)d_in[0];
    const int*   edges = (const int*)d_in[1];
    const float* Wp1 = (const float*)d_in[2];
    const float* bp1 = (const float*)d_in[3];
    const float* Ws1 = (const float*)d_in[4];
    const float* Wn1 = (const float*)d_in[5];
    const float* b1  = (const float*)d_in[6];
    const float* gam = (const float*)d_in[7];
    const float* bet = (const float*)d_in[8];
    const float* Wp2 = (const float*)d_in[9];
    const float* bp2 = (const float*)d_in[10];
    const float* Ws2 = (const float*)d_in[11];
    const float* Wn2 = (const float*)d_in[12];
    const float* b2  = (const float*)d_in[13];
    float* out = (float*)d_out;

    const size_t NF = (size_t)N_NODES * FDIM;
    float* M1   = (float*)d_ws;       // relu(feat@Wp1^T+bp1) — edge-independent
    float* AGG  = M1 + NF;            // segment-max accumulator (int bits)
    float* H    = AGG + NF;           // hidden / bn(hidden)
    float* M2   = H + NF;             // relu(H@Wp2^T+bp2)
    const int nbn = (N_NODES + BN_ROWS - 1) / BN_ROWS;
    float* PART = M2 + NF;            // BN partials [nbn][256]
    float* SC   = PART + (size_t)nbn * 256;  // [256] scale/shift

    const int gemm_grid    = (N_NODES + 127) / 128;
    const int scatter_grid = (N_EDGES + 7) / 8;      // 8 edges per 256-thread block
    const int fill_grid    = 2048;

    // m1 once for all graphs (feat-only)
    dgs_gemm128<<<gemm_grid, 256, 0, stream>>>(feat, Wp1, nullptr, nullptr,
                                               bp1, 1, M1, N_NODES);

    for (int g = 0; g < LOOKBACK; ++g) {
        const int* src = edges + (size_t)g * 2 * N_EDGES;
        const int* dst = src + N_EDGES;

        // conv1 aggregate
        dgs_fill_i32<<<fill_grid, 256, 0, stream>>>((int*)AGG, (long long)NF, INT_MIN);
        dgs_scatter_max<<<scatter_grid, 256, 0, stream>>>(M1, src, dst, (int*)AGG);
        // h1 = feat@Ws1^T + decode(agg)@Wn1^T + b1
        dgs_gemm128<<<gemm_grid, 256, 0, stream>>>(feat, Ws1, AGG, Wn1,
                                                   b1, 0, H, N_NODES);
        // batchnorm + relu (in place)
        dgs_bn_partial<<<nbn, 128, 0, stream>>>(H, PART, N_NODES);
        dgs_bn_finalize<<<1, 128, 0, stream>>>(PART, nbn, gam, bet, SC);
        dgs_bn_apply_relu<<<fill_grid, 256, 0, stream>>>(H, SC, (long long)NF);

        // conv2
        dgs_gemm128<<<gemm_grid, 256, 0, stream>>>(H, Wp2, nullptr, nullptr,
                                                   bp2, 1, M2, N_NODES);
        dgs_fill_i32<<<fill_grid, 256, 0, stream>>>((int*)AGG, (long long)NF, INT_MIN);
        dgs_scatter_max<<<scatter_grid, 256, 0, stream>>>(M2, src, dst, (int*)AGG);
        // out_g = H@Ws2^T + decode(agg)@Wn2^T + b2
        dgs_gemm128<<<gemm_grid, 256, 0, stream>>>(H, Ws2, AGG, Wn2,
                                                   b2, 0, out + (size_t)g * NF, N_NODES);
    }
}